// BilateralFilter_10496900072151
// MI455X (gfx1250) — compile-verified
//
#include <hip/hip_runtime.h>
#include <hip/hip_bf16.h>
#include <stdint.h>

#define IMG   512
#define TILE  32
#define PADW  2
#define S     36            // LDS tile edge (TILE + 2*PADW), also row stride
#define LDS_ELEMS (S * S)   // 1296
#define NL    (-0.72134752044448170368f)   // -0.5 * log2(e)

#if __has_builtin(__builtin_amdgcn_exp2f)
#define EXP2F(x) __builtin_amdgcn_exp2f(x)
#else
#define EXP2F(x) exp2f(x)
#endif

typedef __attribute__((address_space(1))) int g_int;   // global int*
typedef __attribute__((address_space(3))) int l_int;   // LDS int*

// ---- CDNA5 async global->LDS copy (one dword per active lane) ----------------
__device__ __forceinline__ void async_load_dword(uint32_t lds_off, const float* gsrc) {
#if __has_builtin(__builtin_amdgcn_global_load_async_to_lds_b32)
    __builtin_amdgcn_global_load_async_to_lds_b32(
        (g_int*)(uintptr_t)gsrc,
        (l_int*)lds_off,
        0, 0);
#else
    asm volatile("global_load_async_to_lds_b32 %0, %1, off"
                 :: "v"(lds_off), "v"((uint64_t)(uintptr_t)gsrc)
                 : "memory");
#endif
}

__device__ __forceinline__ void wait_async_zero() {
#if __has_builtin(__builtin_amdgcn_s_wait_asynccnt)
    __builtin_amdgcn_s_wait_asynccnt(0);
#else
    asm volatile("s_wait_asynccnt 0" ::: "memory");
#endif
}

// -----------------------------------------------------------------------------
__global__ __launch_bounds__(256)
void BilateralFilter_kernel(const float* __restrict__ X, float* __restrict__ out) {
    __shared__ float tile[LDS_ELEMS];

    const int tid = threadIdx.x;        // 0..255
    const int tx  = tid & 31;           // 0..31 : output column in tile
    const int ty  = tid >> 5;           // 0..7  : row group
    const int gx0 = blockIdx.x * TILE - PADW;
    const int gy0 = blockIdx.y * TILE - PADW;
    const float* __restrict__ Xp = X + (size_t)blockIdx.z * (IMG * IMG);

    // low 32 bits of a generic pointer to LDS == byte offset in LDS aperture
    const uint32_t ldsbase = (uint32_t)(uintptr_t)&tile[0];

    // ---- stage 36x36 halo tile into LDS; zero outside image (zero-pad) ------
    #pragma unroll
    for (int i = 0; i < 6; ++i) {
        const int idx = tid + i * 256;
        if (idx < LDS_ELEMS) {
            const int ly = idx / S;
            const int lx = idx - ly * S;
            const int gy = gy0 + ly;
            const int gx = gx0 + lx;
            if ((unsigned)gy < IMG && (unsigned)gx < IMG) {
                async_load_dword(ldsbase + (uint32_t)idx * 4u, Xp + (gy * IMG + gx));
            } else {
                tile[idx] = 0.0f;
            }
        }
    }
    wait_async_zero();
    __syncthreads();

    // spatial distance^2 table (dy outer, dx inner); NL*const folds at compile time
    const float KD2[5][5] = {
        {8.f, 5.f, 4.f, 5.f, 8.f},
        {5.f, 2.f, 1.f, 2.f, 5.f},
        {4.f, 1.f, 0.f, 1.f, 4.f},
        {5.f, 2.f, 1.f, 2.f, 5.f},
        {8.f, 5.f, 4.f, 5.f, 8.f}};

    // ---- each thread: 4 vertically consecutive output pixels ----------------
    const int y0 = ty * 4;              // local output rows y0..y0+3
    float num[4] = {0.f, 0.f, 0.f, 0.f};
    float den[4] = {0.f, 0.f, 0.f, 0.f};
    float c[4];
    #pragma unroll
    for (int r = 0; r < 4; ++r)
        c[r] = tile[(y0 + r + 2) * S + tx + 2];

    #pragma unroll
    for (int wy = 0; wy < 8; ++wy) {    // LDS rows y0 .. y0+7 feed the 4 outputs
        float v[5];
        #pragma unroll
        for (int j = 0; j < 5; ++j)
            v[j] = tile[(y0 + wy) * S + tx + j];
        #pragma unroll
        for (int r = 0; r < 4; ++r) {
            const int dy = wy - r;      // compile-time per unrolled iteration
            if (dy >= 0 && dy < 5) {
                #pragma unroll
                for (int j = 0; j < 5; ++j) {
                    if (dy == 2 && j == 2) {
                        // center tap: v == c  ->  weight is exactly 1 (skip exp)
                        num[r] += v[j];
                        den[r] += 1.0f;
                    } else {
                        const float t = v[j] - c[r];
                        // w = exp2( NL*(t^2 + d2s) ) == exp(-0.5*(t^2+d2s))
                        const float a = __builtin_fmaf(t * t, NL, NL * KD2[dy][j]);
                        const float e = EXP2F(a);
                        num[r] = __builtin_fmaf(e, v[j], num[r]);
                        den[r] += e;
                    }
                }
            }
        }
    }

    float* __restrict__ Op = out + (size_t)blockIdx.z * (IMG * IMG);
    const int gxo = blockIdx.x * TILE + tx;
    #pragma unroll
    for (int r = 0; r < 4; ++r) {
        const int gyo = blockIdx.y * TILE + y0 + r;
        Op[gyo * IMG + gxo] = num[r] / den[r];
    }
}

// -----------------------------------------------------------------------------
extern "C" void kernel_launch(void* const* d_in, const int* in_sizes, int n_in,
                              void* d_out, int out_size, void* d_ws, size_t ws_size,
                              hipStream_t stream) {
    const float* X = (const float*)d_in[0];
    float* out = (float*)d_out;
    const int planes = in_sizes[0] / (IMG * IMG);   // n*C = 12
    dim3 grid(IMG / TILE, IMG / TILE, planes);
    BilateralFilter_kernel<<<grid, dim3(256), 0, stream>>>(X, out);
}